// CrossAttentionConditioning_8443905704449
// MI455X (gfx1250) — compile-verified
//
#include <hip/hip_runtime.h>
#include <hip/hip_bf16.h>

// ---------------- problem constants ----------------
#define CB   16      // batch
#define CC   768     // channels
#define CD   1024    // cond_dim
#define CS   4096    // H*W
#define TS   32      // spatial positions per LN workgroup
#define LNT  128     // LN block threads

typedef float v2f __attribute__((ext_vector_type(2)));
typedef float v8f __attribute__((ext_vector_type(8)));
typedef float f4  __attribute__((ext_vector_type(4)));   // native float4
typedef int   g4i __attribute__((vector_size(16)));      // int4 (GCC spelling, matches builtin)

#define GLOBAL_AS __attribute__((address_space(1)))
#define LDS_AS    __attribute__((address_space(3)))

// ============================================================
// GEMM: out[16,N] = X[16,K] @ W[N,K]^T + bias[N]
// one wave (32 lanes) per 16-wide column tile, WMMA f32 16x16x4
// ============================================================
__global__ __launch_bounds__(32) void gemm16_wmma(
    const float* __restrict__ X, const float* __restrict__ W,
    const float* __restrict__ bias, float* __restrict__ out,
    int K, int N)
{
    const int n0   = blockIdx.x * 16;
    const int lane = threadIdx.x;       // 0..31
    const int half = lane >> 4;         // 0 | 1
    const int l    = lane & 15;

    const float* __restrict__ xrow = X + (size_t)l * K;          // A row M=l
    const float* __restrict__ wrow = W + (size_t)(n0 + l) * K;   // B col N=n0+l (W row)

    v8f acc = {};
    for (int k0 = 0; k0 < K; k0 += 4) {
        // A 16x4 f32: v0 = K{0|2}, v1 = K{1|3}  (lanes 0-15 -> K0/K1, 16-31 -> K2/K3)
        v2f a, b;
        a.x = xrow[k0 + 2 * half];
        a.y = xrow[k0 + 2 * half + 1];
        // B 4x16 f32: B[k][n] = W[n][k]; same half-split over K
        b.x = wrow[k0 + 2 * half];
        b.y = wrow[k0 + 2 * half + 1];
        acc = __builtin_amdgcn_wmma_f32_16x16x4_f32(
            /*neg_a=*/false, a, /*neg_b=*/false, b,
            /*c_mod=*/(short)0, acc, /*reuse_a=*/false, /*reuse_b=*/false);
    }

    // C/D 16x16 f32: VGPR r -> M=r (lanes 0-15), M=r+8 (lanes 16-31); N = lane&15
    const float bv = bias[n0 + l];
    #pragma unroll
    for (int r = 0; r < 8; ++r) {
        const int m = r + 8 * half;
        out[(size_t)m * N + n0 + l] = acc[r] + bv;
    }
}

// ============================================================
// Fused residual-broadcast + LayerNorm over C, LDS-staged.
// One workgroup: 1 batch element x 32 spatial positions x 768 channels.
// Tile fill uses CDNA5 async global->LDS DMA (ASYNCcnt), so spatial is
// read from HBM exactly once with no VGPR round-trip.
// ============================================================
__global__ __launch_bounds__(LNT) void fused_ln_kernel(
    const float* __restrict__ sp,     // (B,C,S)
    const float* __restrict__ proj,   // (B,C)
    const float* __restrict__ gamma,  // (C)
    const float* __restrict__ beta,   // (C)
    float* __restrict__ out)          // (B,C,S)
{
    extern __shared__ float lds[];
    float* tile = lds;                   // CC*TS   = 24576 floats (96KB)
    float* pch  = tile + CC * TS;        // CC      = 768
    float* red  = pch + CC;              // LNT sums
    float* red2 = red + LNT;             // LNT sumsqs
    float* smu  = red2 + LNT;            // TS
    float* srs  = smu + TS;              // TS

    const int b   = blockIdx.x / (CS / TS);
    const int s0  = (blockIdx.x % (CS / TS)) * TS;
    const int tid = threadIdx.x;

    const float* __restrict__ spb  = sp  + (size_t)b * CC * CS;
    float*       __restrict__ outb = out + (size_t)b * CC * CS;

    // per-batch projection row -> LDS
    for (int c = tid; c < CC; c += LNT) pch[c] = proj[b * CC + c];

    // stage tile: 16B chunks, coalesced (8 chunks per channel row)
    const int NQ = CC * TS / 4;          // 6144 float4s
#if __has_builtin(__builtin_amdgcn_global_load_async_to_lds_b128)
    for (int i = tid; i < NQ; i += LNT) {
        const int c = i >> 3;
        const int q = (i & 7) * 4;
        __builtin_amdgcn_global_load_async_to_lds_b128(
            (GLOBAL_AS g4i*)(spb + (size_t)c * CS + s0 + q),
            (LDS_AS g4i*)(tile + c * TS + q),
            /*imm offset=*/0, /*cpol=*/0);
    }
  #if __has_builtin(__builtin_amdgcn_s_wait_asynccnt)
    __builtin_amdgcn_s_wait_asynccnt(0);
  #else
    asm volatile("s_wait_asynccnt 0x0" ::: "memory");
  #endif
#else
    for (int i = tid; i < NQ; i += LNT) {
        const int c = i >> 3;
        const int q = (i & 7) * 4;
        const f4 v = *(const f4*)(spb + (size_t)c * CS + s0 + q);
        *(f4*)(tile + c * TS + q) = v;
    }
#endif
    __syncthreads();

    // partial reduction: 4 thread-groups per spatial position
    const int sl   = tid & 31;           // spatial position in tile
    const int part = tid >> 5;           // 0..3
    float sum = 0.f, ssq = 0.f;
    for (int c = part; c < CC; c += 4) {
        const float x = tile[c * TS + sl] + pch[c];
        sum += x;
        ssq += x * x;
    }
    red [part * TS + sl] = sum;
    red2[part * TS + sl] = ssq;
    __syncthreads();

    if (part == 0) {
        const float t  = red [sl] + red [TS + sl] + red [2 * TS + sl] + red [3 * TS + sl];
        const float t2 = red2[sl] + red2[TS + sl] + red2[2 * TS + sl] + red2[3 * TS + sl];
        const float mu  = t * (1.0f / CC);
        const float var = t2 * (1.0f / CC) - mu * mu;
        smu[sl] = mu;
        srs[sl] = rsqrtf(var + 1e-5f);
    }
    __syncthreads();

    // normalize from LDS, coalesced non-temporal b128 stores (output is write-once)
    for (int i = tid; i < NQ; i += LNT) {
        const int c = i >> 3;
        const int q = (i & 7) * 4;
        const f4 v = *(const f4*)(tile + c * TS + q);
        const f4 m = *(const f4*)(smu + q);
        const f4 r = *(const f4*)(srs + q);
        const float g = gamma[c], bt = beta[c], p = pch[c];
        f4 y = ((v + p) - m) * r * g + bt;
        __builtin_nontemporal_store(y, (f4*)(outb + (size_t)c * CS + s0 + q));
    }
}

// ============================================================
extern "C" void kernel_launch(void* const* d_in, const int* in_sizes, int n_in,
                              void* d_out, int out_size, void* d_ws, size_t ws_size,
                              hipStream_t stream) {
    const float* spatial      = (const float*)d_in[0];   // (16,768,64,64)
    const float* conditioning = (const float*)d_in[1];   // (16,1024)
    const float* w_cond       = (const float*)d_in[2];   // (768,1024)
    const float* b_cond       = (const float*)d_in[3];   // (768)
    const float* in_proj_w    = (const float*)d_in[4];   // (2304,768)
    const float* in_proj_b    = (const float*)d_in[5];   // (2304)
    const float* attn_out_w   = (const float*)d_in[6];   // (768,768)
    const float* attn_out_b   = (const float*)d_in[7];   // (768)
    const float* w_out        = (const float*)d_in[8];   // (768,768)
    const float* b_out        = (const float*)d_in[9];   // (768)
    const float* ln_gamma     = (const float*)d_in[10];  // (768)
    const float* ln_beta      = (const float*)d_in[11];  // (768)
    float* out = (float*)d_out;

    // ping-pong workspace buffers for the 4-GEMM chain (16*768 floats each)
    float* buf0 = (float*)d_ws;
    float* buf1 = buf0 + CB * CC;

    const float* wv = in_proj_w + (size_t)2 * CC * CC;   // V projection rows
    const float* bv = in_proj_b + 2 * CC;

    dim3 gblk(CC / 16);  // 48 waves
    // cond = conditioning @ w_cond^T + b_cond
    gemm16_wmma<<<gblk, 32, 0, stream>>>(conditioning, w_cond, b_cond, buf0, CD, CC);
    // v = cond @ wv^T + bv
    gemm16_wmma<<<gblk, 32, 0, stream>>>(buf0, wv, bv, buf1, CC, CC);
    // attn = v @ attn_out_w^T + attn_out_b
    gemm16_wmma<<<gblk, 32, 0, stream>>>(buf1, attn_out_w, attn_out_b, buf0, CC, CC);
    // proj = attn @ w_out^T + b_out
    gemm16_wmma<<<gblk, 32, 0, stream>>>(buf0, w_out, b_out, buf1, CC, CC);

    // fused residual + LayerNorm, ~100KB LDS tile per workgroup (CDNA5: 320KB/WGP)
    const size_t lds_bytes = (size_t)(CC * TS + CC + 2 * LNT + 2 * TS) * sizeof(float);
    fused_ln_kernel<<<dim3(CB * (CS / TS)), LNT, lds_bytes, stream>>>(
        spatial, buf1, ln_gamma, ln_beta, out);
}